// Word2MatEncoder_75617194213385
// MI455X (gfx1250) — compile-verified
//
#include <hip/hip_runtime.h>
#include <stdint.h>

// CDNA5 / gfx1250, wave32. Word2Mat chain: out[b] = prod_s table[sent[b,s]] (28x28 fp32).
// fp32-exact path: V_WMMA_F32_16X16X4_F32; table gather via GLOBAL_LOAD_ASYNC_TO_LDS_B32.

typedef __attribute__((ext_vector_type(2))) float v2f;
typedef __attribute__((ext_vector_type(8))) float v8f;

#define D28    28
#define EMB    784
#define SEQ    64
#define PADN   32
#define STRIDE 34   // LDS row stride (floats): float2 stays 8B aligned, halves hit disjoint banks

__global__ __launch_bounds__(128) void w2m_chain_kernel(
    const float* __restrict__ table,
    const int*   __restrict__ sent,
    float*       __restrict__ out)
{
    __shared__ float curb[2][PADN * STRIDE];  // running product, row-major, double-buffered
    __shared__ float nxt[2][PADN * STRIDE];   // next matrix, TRANSPOSED (Mt[n][k]), double-buffered

    const int b = blockIdx.x;
    const int t = threadIdx.x;          // 0..127 (4 wave32)
    __builtin_assume(t < 128 && t >= 0);
    const int wave  = t >> 5;
    const int lane  = t & 31;
    const int mi    = wave >> 1;        // output tile row (0..1)
    const int ni    = wave & 1;         // output tile col (0..1)
    const int lmod  = lane & 15;
    const int lhalf = lane >> 4;        // 0 | 1

    const int* sidx = sent + b * SEQ;

    // ---- zero pads once (zero-padding is product-invariant) ----
    for (int i = t; i < PADN * STRIDE; i += 128) {
        curb[0][i] = 0.0f; curb[1][i] = 0.0f;
        nxt[0][i]  = 0.0f; nxt[1][i]  = 0.0f;
    }

    // ---- per-thread copy offsets, invariant across steps (clamped tail: no divergence) ----
    unsigned srcoff[7];   // byte offset within a table row
    unsigned dstrel[7];   // byte offset within an nxt buffer (transposed slot)
    #pragma unroll
    for (int i = 0; i < 7; ++i) {
        int e  = t + i * 128;
        int ec = e < EMB ? e : (EMB - 1);   // dup lanes rewrite the same value: benign
        int r = ec / D28, c = ec - r * D28;
        srcoff[i] = (unsigned)ec * 4u;
        dstrel[i] = (unsigned)(c * STRIDE + r) * 4u;
    }
    __syncthreads();

    // ---- curb[0] = M_0 (row-major); nxt[1] = M_1 (transposed) ----
    {
        const float* r0 = table + (long)sidx[0] * EMB;
        const float* r1 = table + (long)sidx[1] * EMB;
        #pragma unroll
        for (int i = 0; i < 7; ++i) {
            int e  = t + i * 128;
            int ec = e < EMB ? e : (EMB - 1);
            int r = ec / D28, c = ec - r * D28;
            curb[0][r * STRIDE + c] = r0[ec];
            nxt[1][c * STRIDE + r]  = r1[ec];
        }
    }
    __syncthreads();

    const int arow = (mi * 16 + lmod) * STRIDE;
    const int brow = (ni * 16 + lmod) * STRIDE;
    const int koff = lhalf * 2;

    for (int s = 1; s < SEQ; ++s) {
        const int rb = (s + 1) & 1;     // cur read buffer / nxt prefetch target
        const int wb = s & 1;           // cur write buffer / nxt read buffer

        // -------- async prefetch M_{s+1} -> nxt[rb], scatter-transposed into LDS --------
        if (s + 1 < SEQ) {
            const unsigned long long rn =
                (unsigned long long)(uintptr_t)(table + (long)sidx[s + 1] * EMB);
            const unsigned nbase = (unsigned)(uintptr_t)(&nxt[rb][0]);
            #pragma unroll
            for (int i = 0; i < 7; ++i) {
                unsigned dst = nbase + dstrel[i];
                asm volatile("global_load_async_to_lds_b32 %0, %1, %2"
                             :: "v"(dst), "v"(srcoff[i]), "s"(rn) : "memory");
            }
        }

        // -------- acc(16x16) = [cur @ M_s] tile (mi,ni); K=32 in 8 chunks of 4 --------
        const float* A  = curb[rb];
        const float* Bt = nxt[wb];
        v8f acc = {};
        #pragma unroll
        for (int kc = 0; kc < 8; ++kc) {
            const int kb = kc * 4 + koff;
            v2f a  = *(const v2f*)&A[arow + kb];
            v2f bb = *(const v2f*)&Bt[brow + kb];
            acc = __builtin_amdgcn_wmma_f32_16x16x4_f32(
                      false, a, false, bb, (short)0, acc, false, false);
        }

        // -------- commit D tile into the other cur buffer (C/D layout: VGPR v -> rows v, v+8) --------
        float* Cw = curb[wb];
        #pragma unroll
        for (int v = 0; v < 8; ++v)
            Cw[(mi * 16 + v + lhalf * 8) * STRIDE + ni * 16 + lmod] = acc[v];

        // async writes must land before anyone reads nxt[rb] next step
        asm volatile("s_wait_asynccnt 0" ::: "memory");
        __syncthreads();   // single barrier per step (cur is double-buffered)
    }

    // -------- write result (un-pad), coalesced --------
    const float* P = curb[(SEQ - 1) & 1];
    float* ob = out + (long)b * EMB;
    #pragma unroll
    for (int i = 0; i < 7; ++i) {
        int e = t + i * 128;
        if (e < EMB) {
            int r = e / D28, c = e - r * D28;
            ob[e] = P[r * STRIDE + c];
        }
    }
}

extern "C" void kernel_launch(void* const* d_in, const int* in_sizes, int n_in,
                              void* d_out, int out_size, void* d_ws, size_t ws_size,
                              hipStream_t stream) {
    const float* table = (const float*)d_in[0];   // (32001, 784) fp32
    const int*   sent  = (const int*)d_in[1];     // (2048, 64) int32
    float*       out   = (float*)d_out;           // (2048, 784) fp32
    (void)d_ws; (void)ws_size; (void)n_in; (void)out_size;

    const int B = in_sizes[1] / SEQ;              // 2048
    hipLaunchKernelGGL(w2m_chain_kernel, dim3(B), dim3(128), 0, stream,
                       table, sent, out);
}